// Trajectory_89893665505806
// MI455X (gfx1250) — compile-verified
//
#include <hip/hip_runtime.h>
#include <math.h>

typedef float v2f __attribute__((ext_vector_type(2)));
typedef float v8f __attribute__((ext_vector_type(8)));
typedef unsigned int u32x4 __attribute__((ext_vector_type(4)));
typedef int i32x4 __attribute__((ext_vector_type(4)));
typedef int i32x8 __attribute__((ext_vector_type(8)));

#define MBLK 128
#define NBLK 128
#define KBLK 32

#if __has_builtin(__builtin_amdgcn_tensor_load_to_lds) && __has_builtin(__builtin_amdgcn_s_wait_tensorcnt)
#define USE_TDM 1
#else
#define USE_TDM 0
#endif

// Compiler fence: the TDM writes LDS behind the compiler's back. Taking the
// LDS pointer as an asm operand with a memory clobber makes it a potential
// writer, so ds_loads from smem can no longer be folded to undef.
#define LDS_PUBLISH(p) asm volatile("" : : "r"(p) : "memory")

// LDS layout (floats): [A0:128x33][B0:32x129][A1:128x33][B1:32x129]
#define A_TILE_F (MBLK * (KBLK + 1))          // 4224
#define B_TILE_F (KBLK * (NBLK + 1))          // 4128
#define BUF_F (A_TILE_F + B_TILE_F)           // 8352
#define A_OFF_BYTES(d) ((unsigned)(d) * BUF_F * 4u)
#define B_OFF_BYTES(d) ((unsigned)((d) * BUF_F + A_TILE_F) * 4u)

#if USE_TDM
// 2D TDM load: global (row-major, stride0 elems of 4B) -> LDS with 1-dword row pad.
// tensor dims = remaining extent from tile origin (OOB reads return zero -> edge tiles).
__device__ __forceinline__ void tdm_load_2d(
    unsigned lds_byte_off, const float* gptr,
    unsigned tensor_d0, unsigned tensor_d1,
    unsigned tile_d0, unsigned tile_d1,
    unsigned long long stride0, unsigned pad_interval_code)
{
  const unsigned long long ga = (unsigned long long)gptr;
  u32x4 g0;
  g0[0] = 1u;                                           // count=1, user-mode D#
  g0[1] = lds_byte_off;                                 // lds_addr
  g0[2] = (unsigned)(ga & 0xffffffffull);               // global_addr[31:0]
  g0[3] = (unsigned)((ga >> 32) & 0x01ffffffull)        // global_addr[56:32]
        | (2u << 30);                                   // type = 2 (image)
  i32x8 g1;
  g1[0] = (int)((2u << 16)                              // data_size = 4B
        | (1u << 20)                                    // pad_enable
        | (pad_interval_code << 22)                     // rows of 2^(c+1) dwords
        | (0u << 25));                                  // pad_amount = 1 dword
  g1[1] = (int)((tensor_d0 & 0xffffu) << 16);           // [63:48] tensor_dim0 lo16
  g1[2] = (int)((tensor_d0 >> 16) | ((tensor_d1 & 0xffffu) << 16));
  g1[3] = (int)((tensor_d1 >> 16) | (tile_d0 << 16));   // [127:112] tile_dim0
  g1[4] = (int)(tile_d1 & 0xffffu);                     // tile_dim1 (tile_dim2 = 0)
  g1[5] = (int)(unsigned)(stride0 & 0xffffffffull);     // tensor_dim0_stride lo32
  g1[6] = (int)(unsigned)((stride0 >> 32) & 0xffffull); // stride hi16 (stride1 = 0)
  g1[7] = 0;
  i32x4 z4 = {0, 0, 0, 0};
#if defined(__clang_major__) && (__clang_major__ >= 23)
  i32x8 z8 = {0, 0, 0, 0, 0, 0, 0, 0};
  __builtin_amdgcn_tensor_load_to_lds(g0, g1, z4, z4, z8, 0);
#else
  __builtin_amdgcn_tensor_load_to_lds(g0, g1, z4, z4, 0);
#endif
}
#endif

// C[M,N] = A[M,K] * B[K,N], fp32 via V_WMMA_F32_16X16X4_F32.
// 256 threads = 8 wave32 arranged 2(M) x 4(N); each wave owns 64x32 = 4x2 frags.
// Tiles staged by the Tensor Data Mover into double-buffered LDS; DMA of tile
// t+1 overlaps WMMA on tile t (s_wait_tensorcnt 2 keeps one tile in flight).
__global__ __launch_bounds__(256) void gemm_f32_wmma(
    const float* __restrict__ A, int lda,
    const float* __restrict__ B, int ldb,
    float* __restrict__ C, int ldc,
    int M, int N, int K)
{
  __shared__ float smem[2 * BUF_F];

  const int tid  = threadIdx.x;
  const int lane = tid & 31;
  const int wave = tid >> 5;
  const int wm   = wave >> 2;            // 0..1 -> 64-row group
  const int wn   = wave & 3;             // 0..3 -> 32-col group

  const int m0 = blockIdx.y * MBLK;
  const int n0 = blockIdx.x * NBLK;

  v8f acc[4][2];
  #pragma unroll
  for (int i = 0; i < 4; ++i)
    #pragma unroll
    for (int j = 0; j < 2; ++j)
      acc[i][j] = (v8f)0.0f;

  const int frag16 = lane & 15;          // M (A-frag) / N (B,C-frag)
  const int khalf  = (lane >> 4) << 1;   // K pair: 0 or 2
  const int mhalf  = (lane >> 4) << 3;   // C/D row offset: 0 or 8

  const int T = (K + KBLK - 1) / KBLK;

#if USE_TDM
  if (tid < 32) {   // wave 0 drives the TDM (execz-skip for waves 1..7)
    tdm_load_2d(A_OFF_BYTES(0), A + (size_t)m0 * lda,
                (unsigned)K, (unsigned)(M - m0), KBLK, MBLK,
                (unsigned long long)lda, 4);   // 32-dword rows -> pad code 4
    tdm_load_2d(B_OFF_BYTES(0), B + n0,
                (unsigned)(N - n0), (unsigned)K, NBLK, KBLK,
                (unsigned long long)ldb, 6);   // 128-dword rows -> pad code 6
  }
#endif

  for (int t = 0; t < T; ++t) {
    const int kk = t * KBLK;
    const int d  = t & 1;
    float* sAd = smem + d * BUF_F;                  // [128][33]
    float* sBd = smem + d * BUF_F + A_TILE_F;       // [32][129]

#if USE_TDM
    if (tid < 32) {
      if (t + 1 < T) {
        const int nk = kk + KBLK;
        const int nd = d ^ 1;
        tdm_load_2d(A_OFF_BYTES(nd), A + (size_t)m0 * lda + nk,
                    (unsigned)(K - nk), (unsigned)(M - m0), KBLK, MBLK,
                    (unsigned long long)lda, 4);
        tdm_load_2d(B_OFF_BYTES(nd), B + (size_t)nk * ldb + n0,
                    (unsigned)(N - n0), (unsigned)(K - nk), NBLK, KBLK,
                    (unsigned long long)ldb, 6);
        __builtin_amdgcn_s_wait_tensorcnt(2);       // tile t complete
      } else {
        __builtin_amdgcn_s_wait_tensorcnt(0);
      }
    }
#else
    { // manual staging fallback
      const int ar = tid >> 1, ac0 = (tid & 1) * 16;
      #pragma unroll
      for (int c = 0; c < 16; ++c) {
        const int gm = m0 + ar, gk = kk + ac0 + c;
        sAd[ar * (KBLK + 1) + ac0 + c] =
            (gm < M && gk < K) ? A[(size_t)gm * lda + gk] : 0.0f;
      }
      const int br = tid >> 3, bc0 = (tid & 7) * 16;
      #pragma unroll
      for (int c = 0; c < 16; ++c) {
        const int gk = kk + br, gn = n0 + bc0 + c;
        sBd[br * (NBLK + 1) + bc0 + c] =
            (gk < K && gn < N) ? B[(size_t)gk * ldb + gn] : 0.0f;
      }
    }
#endif
    __syncthreads();
    LDS_PUBLISH(smem);   // TDM wrote LDS: pin the ds_loads below

    // 8 k-steps of K=4; 8 WMMAs per step per wave
    #pragma unroll
    for (int k4 = 0; k4 < KBLK; k4 += 4) {
      v2f a[4], b[2];
      #pragma unroll
      for (int i = 0; i < 4; ++i) {
        const float* p = &sAd[(wm * 64 + i * 16 + frag16) * (KBLK + 1) + k4 + khalf];
        a[i].x = p[0]; a[i].y = p[1];
      }
      #pragma unroll
      for (int j = 0; j < 2; ++j) {
        const int col = wn * 32 + j * 16 + frag16;
        b[j].x = sBd[(k4 + khalf) * (NBLK + 1) + col];
        b[j].y = sBd[(k4 + khalf + 1) * (NBLK + 1) + col];
      }
      #pragma unroll
      for (int i = 0; i < 4; ++i)
        #pragma unroll
        for (int j = 0; j < 2; ++j)
          acc[i][j] = __builtin_amdgcn_wmma_f32_16x16x4_f32(
              false, a[i], false, b[j], (short)0, acc[i][j], false, false);
    }
    __syncthreads();
  }

  // D layout: M = e + 8*lane[4], N = lane[3:0]; adjacent lanes -> coalesced
  #pragma unroll
  for (int i = 0; i < 4; ++i) {
    #pragma unroll
    for (int j = 0; j < 2; ++j) {
      #pragma unroll
      for (int e = 0; e < 8; ++e) {
        const int m = m0 + wm * 64 + i * 16 + mhalf + e;
        const int n = n0 + wn * 32 + j * 16 + frag16;
        if (m < M && n < N)
          __builtin_nontemporal_store(acc[i][j][e], &C[(size_t)m * ldc + n]);
      }
    }
  }
}

// E = [relu(z) | relu(z_a) | ones | zeros] (8000 x 144); also emit hiden_emb = z.
__global__ void build_E(const float* __restrict__ zz, float* __restrict__ E,
                        float* __restrict__ hiden, int Nrows)
{
  const int idx = blockIdx.x * blockDim.x + threadIdx.x;
  if (idx >= Nrows * 144) return;
  const int r = idx / 144, c = idx % 144;
  float v;
  if (c < 128) {
    const float z = zz[(size_t)r * 128 + c];
    v = z > 0.0f ? z : 0.0f;
    if (c < 64) hiden[(size_t)r * 64 + c] = z;
  } else {
    v = (c == 128) ? 1.0f : 0.0f;       // ones column -> rowsum via the GEMM
  }
  E[idx] = v;
}

// Per-row readout + bilinear discriminator. Block = 64 threads (2 waves), 1 row/block.
__global__ __launch_bounds__(64) void readout_disc(
    const float* __restrict__ GM, const float* __restrict__ E,
    const float* __restrict__ W, const float* __restrict__ bptr,
    float* __restrict__ ret, float* __restrict__ reta,
    float* __restrict__ loss, int Nrows)
{
  __shared__ float sEmb[64], sEmbA[64], sG[64], sGA[64], red[64];
  const int r = blockIdx.x;
  const int t = threadIdx.x;

  const float rs = GM[(size_t)r * 144 + 128];
  const float u  = GM[(size_t)r * 144 + t]      / rs;
  const float ua = GM[(size_t)r * 144 + 64 + t] / rs;

  red[t] = u * u;  __syncthreads();
  for (int s = 32; s > 0; s >>= 1) { if (t < s) red[t] += red[t + s]; __syncthreads(); }
  const float n1 = fmaxf(sqrtf(red[0]), 1e-12f); __syncthreads();
  red[t] = ua * ua; __syncthreads();
  for (int s = 32; s > 0; s >>= 1) { if (t < s) red[t] += red[t + s]; __syncthreads(); }
  const float n2 = fmaxf(sqrtf(red[0]), 1e-12f); __syncthreads();

  sG[t]  = 1.0f / (1.0f + __expf(-(u  / n1)));
  sGA[t] = 1.0f / (1.0f + __expf(-(ua / n2)));
  sEmb[t]  = E[(size_t)r * 144 + t];
  sEmbA[t] = E[(size_t)r * 144 + 64 + t];
  __syncthreads();

  float hw = 0.0f, hwa = 0.0f;
  #pragma unroll 8
  for (int k = 0; k < 64; ++k) {
    const float w = W[k * 64 + t];
    hw  += sEmb[k]  * w;
    hwa += sEmbA[k] * w;
  }

  const float b = bptr[0];
  float sums[4];
  const float parts[4] = { hw * sG[t], hwa * sG[t], hwa * sGA[t], hw * sGA[t] };
  #pragma unroll
  for (int q = 0; q < 4; ++q) {
    red[t] = parts[q]; __syncthreads();
    for (int s = 32; s > 0; s >>= 1) { if (t < s) red[t] += red[t + s]; __syncthreads(); }
    sums[q] = red[0]; __syncthreads();
  }

  if (t == 0) {
    ret[(size_t)r * 2 + 0]  = sums[0] + b;
    ret[(size_t)r * 2 + 1]  = sums[1] + b;
    reta[(size_t)r * 2 + 0] = sums[2] + b;
    reta[(size_t)r * 2 + 1] = sums[3] + b;
    if (r == 0) loss[0] = 0.0f;
  }
}

extern "C" void kernel_launch(void* const* d_in, const int* in_sizes, int n_in,
                              void* d_out, int out_size, void* d_ws, size_t ws_size,
                              hipStream_t stream) {
  const int N = 8000, DIN = 1024, DH = 64;
  const float* feat  = (const float*)d_in[0];
  const float* feata = (const float*)d_in[1];
  const float* adj   = (const float*)d_in[2];
  const float* gn    = (const float*)d_in[3];
  const float* w1    = (const float*)d_in[4];
  const float* w2    = (const float*)d_in[5];
  const float* dw    = (const float*)d_in[6];
  const float* db    = (const float*)d_in[7];

  float* out   = (float*)d_out;
  float* hiden = out;                               // [N, 64]
  float* h     = out + (size_t)N * DH;              // [N, 1024]
  float* ret   = h   + (size_t)N * DIN;             // [N, 2]
  float* reta  = ret + (size_t)N * 2;               // [N, 2]
  float* loss  = reta + (size_t)N * 2;              // scalar

  float* ws = (float*)d_ws;
  float* B1 = ws;                                   // [N, 128] = [t1 | t3]
  float* ZZ = B1 + (size_t)N * 128;                 // [N, 128] = adj @ B1
  float* E  = ZZ + (size_t)N * 128;                 // [N, 144]
  float* T2 = E  + (size_t)N * 144;                 // [N, 1024] = z @ W2
  float* GM = T2 + (size_t)N * DIN;                 // [N, 144] = gn @ E

  const dim3 blk(256);
  auto grid = [](int n, int m) { return dim3((n + NBLK - 1) / NBLK, (m + MBLK - 1) / MBLK); };

  // t1 = feat @ W1 ; t3 = feat_a @ W1  (packed side-by-side in B1)
  gemm_f32_wmma<<<grid(DH, N),  blk, 0, stream>>>(feat,  DIN, w1, DH,  B1,      128, N, DH,  DIN);
  gemm_f32_wmma<<<grid(DH, N),  blk, 0, stream>>>(feata, DIN, w1, DH,  B1 + DH, 128, N, DH,  DIN);
  // [z | z_a] = adj @ [t1 | t3]   (adj streamed once for both)
  gemm_f32_wmma<<<grid(128, N), blk, 0, stream>>>(adj,   N,   B1, 128, ZZ,      128, N, 128, N);
  build_E<<<(N * 144 + 255) / 256, 256, 0, stream>>>(ZZ, E, hiden, N);
  // t2 = z @ W2   (z = ZZ[:, :64], lda = 128)
  gemm_f32_wmma<<<grid(DIN, N), blk, 0, stream>>>(ZZ,    128, w2, DIN, T2,      DIN, N, DIN, DH);
  // h = adj @ t2  (the 131 GF GEMM)
  gemm_f32_wmma<<<grid(DIN, N), blk, 0, stream>>>(adj,   N,   T2, DIN, h,       DIN, N, DIN, N);
  // GM = graph_neigh @ E  -> readouts + rowsum (ones column) in one pass
  gemm_f32_wmma<<<grid(144, N), blk, 0, stream>>>(gn,    N,   E,  144, GM,      144, N, 144, N);
  readout_disc<<<N, 64, 0, stream>>>(GM, E, dw, db, ret, reta, loss, N);
}